// MinFinder_6090263626339
// MI455X (gfx1250) — compile-verified
//
#include <hip/hip_runtime.h>

// Input order assumption: setup_inputs() dict order, params flattened in
// insertion order:
//  0:x 1:last_edge 2:embed_W 3:embed_b
//  enc: 4:wq 5:wk 6:ab 7:ff1_W 8:ff1_b 9:ff2_W 10:ff2_b 11:ln1_g 12:ln1_b 13:ln2_g 14:ln2_b
//  dec: 15:wq 16:wk ... (rest of decoder params unused — mathematically dead:
//  the returned cross-attn weights are softmax_j(h_enc . dec.wk[5]) masked.)

typedef float v2f __attribute__((ext_vector_type(2)));
typedef float v8f __attribute__((ext_vector_type(8)));

#define NEGVAL (-1e30f)
#define LNEPS  (1e-5f)

constexpr int B_ = 4, N_ = 2048, D_ = 64, FF_ = 256, L_ = 6;

// FFN tile config: 64 tokens per block, 4 waves (16 rows each).
constexpr int TOK_PER_BLK = 64;
constexpr int FFN_THREADS = 128;
// Dynamic LDS layout (floats):
constexpr int WSH_OFF = 0;                     // staged W1 / W2 panel: 64*256
constexpr int ALN_OFF = WSH_OFF + D_ * FF_;    // LN1 activations: 64*64
constexpr int MID_OFF = ALN_OFF + TOK_PER_BLK * D_;  // relu mid: 64*256
constexpr int FFN_LDS_FLOATS = MID_OFF + TOK_PER_BLK * FF_;
constexpr size_t FFN_LDS_BYTES = (size_t)FFN_LDS_FLOATS * sizeof(float); // 144KB

// ---- cooperative W-panel staging: 16384 floats (64KB) via async-to-LDS ----
#if defined(__has_builtin)
#if __has_builtin(__builtin_amdgcn_global_load_async_to_lds_b128)
#define HAVE_ASYNC_LDS 1
#endif
#endif

#ifdef HAVE_ASYNC_LDS
// Builtin signature (from hipcc diagnostic): param0 is a pointer to
// vector_size(16) int in the global (AS1 / "__device__") address space;
// param1 is the LDS (AS3) counterpart; then imm offset and cpol.
typedef int ai4 __attribute__((vector_size(16)));
typedef __attribute__((address_space(1))) ai4* gas_ai4p;
typedef __attribute__((address_space(3))) ai4* las_ai4p;
#endif

__device__ __forceinline__ void stage_w_panel(const float* __restrict__ g,
                                              float* lds, int tid) {
#ifdef HAVE_ASYNC_LDS
  // 16384 floats = 4096 x 16B chunks; 128 threads -> 32 chunks each.
  for (int c = tid; c < (D_ * FF_) / 4; c += FFN_THREADS) {
    __builtin_amdgcn_global_load_async_to_lds_b128(
        (gas_ai4p)(g + (size_t)c * 4),
        (las_ai4p)(lds + (size_t)c * 4),
        /*imm offset*/ 0, /*cpol*/ 0);
  }
  asm volatile("s_wait_asynccnt 0" ::: "memory");
#else
  const float4* gv = (const float4*)g;
  float4* lv = (float4*)lds;
  for (int c = tid; c < (D_ * FF_) / 4; c += FFN_THREADS) lv[c] = gv[c];
#endif
}

// ---------------------------------------------------------------- masks
__global__ void masks_kernel(const float* __restrict__ x,
                             float* __restrict__ encM,
                             float* __restrict__ memM) {
  const int b = blockIdx.x, tid = threadIdx.x;
  __shared__ int badsh;
  if (tid == 0) badsh = 0;
  __syncthreads();
  int lb = 0;
  for (int j = tid; j < N_; j += blockDim.x) {
    float x0 = x[((size_t)b * N_ + j) * 2 + 0];
    float x1 = x[((size_t)b * N_ + j) * 2 + 1];
    bool fake = !(x0 == 0.f && x1 == 0.f);
    bool vis  = (x0 == -1.f && x1 == -1.f);
    if (!fake && vis) lb = 1;
  }
  if (lb) atomicOr(&badsh, 1);
  __syncthreads();
  const bool bad = (badsh != 0);
  for (int j = tid; j < N_; j += blockDim.x) {
    float x0 = x[((size_t)b * N_ + j) * 2 + 0];
    float x1 = x[((size_t)b * N_ + j) * 2 + 1];
    bool fake = !(x0 == 0.f && x1 == 0.f);
    bool vis  = (x0 == -1.f && x1 == -1.f);
    if (j == 0) { fake = fake || bad; vis = vis || bad; }
    encM[(size_t)b * N_ + j] = (!fake && vis) ? 1.f : 0.f;  // encoder key mask
    memM[(size_t)b * N_ + j] = (!fake) ? 1.f : 0.f;         // memory key mask
  }
}

// ---------------------------------------------------------------- embed
__global__ void embed_kernel(const float* __restrict__ x,
                             const float* __restrict__ last,
                             const float* __restrict__ eW,   // [3,64]
                             const float* __restrict__ eb,   // [64]
                             float* __restrict__ h) {
  int i = blockIdx.x * blockDim.x + threadIdx.x;
  if (i >= B_ * N_ * D_) return;
  int d = i & (D_ - 1);
  int bn = i >> 6;
  float x0 = x[(size_t)bn * 2 + 0];
  float x1 = x[(size_t)bn * 2 + 1];
  float le = last[bn];
  h[i] = x0 * eW[d] + x1 * eW[D_ + d] + le * eW[2 * D_ + d] + eb[d];
}

// ------------------------------------------- rank-1 attention softmax (+ctx)
// logits s[b,i,j] = u_i + v_j + ab: softmax over keys j is invariant to the
// per-row constant, so weights are query-independent. 1024 threads/block.
// WRITE_W=false: ctx[b,64] = sum_j a_j h[b,j,:]   (encoder layer)
// WRITE_W=true : write weights a[b,j] to outw     (final output)
template <bool WRITE_W>
__global__ void __launch_bounds__(1024)
attn_ctx_kernel(const float* __restrict__ h,
                const float* __restrict__ wk,    // [64]
                const float* __restrict__ maskF, // [B,N] 1=masked
                float* __restrict__ ctx,
                float* __restrict__ outw) {
  __shared__ float vsh[N_];
  __shared__ float red[1024];
  __shared__ float wksh[D_];
  __shared__ float part[1024];
  const int b = blockIdx.x, tid = threadIdx.x;
  const float* hb = h + (size_t)b * N_ * D_;
  if (tid < D_) wksh[tid] = wk[tid];
  __syncthreads();

  float lmax = -3.4e38f;
  for (int j = tid; j < N_; j += 1024) {
    float v = 0.f;
    const float* hr = hb + (size_t)j * D_;
#pragma unroll
    for (int d = 0; d < D_; ++d) v += hr[d] * wksh[d];
    if (maskF[(size_t)b * N_ + j] != 0.f) v = NEGVAL;
    vsh[j] = v;
    lmax = fmaxf(lmax, v);
  }
  red[tid] = lmax;
  __syncthreads();
  for (int s = 512; s > 0; s >>= 1) {
    if (tid < s) red[tid] = fmaxf(red[tid], red[tid + s]);
    __syncthreads();
  }
  const float mx = red[0];
  __syncthreads();
  float lsum = 0.f;
  for (int j = tid; j < N_; j += 1024) {
    float e = expf(vsh[j] - mx);   // masked -> exp(-1e30) == 0, as reference
    vsh[j] = e;
    lsum += e;
  }
  red[tid] = lsum;
  __syncthreads();
  for (int s = 512; s > 0; s >>= 1) {
    if (tid < s) red[tid] += red[tid + s];
    __syncthreads();
  }
  const float invZ = 1.f / red[0];

  if (WRITE_W) {
    for (int j = tid; j < N_; j += 1024)
      outw[(size_t)b * N_ + j] = vsh[j] * invZ;
  } else {
    const int d = tid & (D_ - 1);
    const int chunk = tid >> 6;  // 0..15, 128 tokens each
    float p = 0.f;
    for (int j = chunk * (N_ / 16); j < (chunk + 1) * (N_ / 16); ++j)
      p += vsh[j] * hb[(size_t)j * D_ + d];
    part[tid] = p;
    __syncthreads();
    if (tid < D_) {
      float acc = 0.f;
#pragma unroll
      for (int c = 0; c < 16; ++c) acc += part[c * 64 + tid];
      ctx[b * D_ + tid] = acc * invZ;
    }
  }
}

// ------------------------------------------- fused LN1 + FFN (WMMA) + LN2
// 4 waves x 16-token tiles per block; W1/W2 staged once per block into a
// shared LDS panel (async global->LDS when available) and reused by all 4
// waves: 4x less L2 traffic for weights. fp32 WMMA 16x16x4 keeps reference
// precision. Layouts per ISA 7.12.2:
//   A 16x4 f32: lane l -> row m=l%16, k-pair kbase + (l/16)*2 in {v0,v1}
//   B 4x16 f32: lane l -> col n=l%16, same k-pair split
//   C/D 16x16 : VGPR i -> M = i + (l/16)*8, N = l%16
__global__ void __launch_bounds__(FFN_THREADS)
enc_ffn_kernel(float* __restrict__ h, const float* __restrict__ ctx,
               const float* __restrict__ W1, const float* __restrict__ b1,
               const float* __restrict__ W2, const float* __restrict__ b2,
               const float* __restrict__ g1, const float* __restrict__ be1,
               const float* __restrict__ g2, const float* __restrict__ be2) {
  extern __shared__ float smem[];
  float* Wsh = smem + WSH_OFF;  // [64][256] (W1) then [256][64] (W2)
  float* Aln = smem + ALN_OFF;  // [64][64]  LN1(h+ctx), later += FFN out
  float* Mid = smem + MID_OFF;  // [64][256] relu(Aln @ W1 + b1)

  const int tid = threadIdx.x;
  const int lane = tid & 31;
  const int wave = tid >> 5;
  const int wr0 = wave * 16;  // this wave's first row within the block tile
  const int blocksPerBatch = N_ / TOK_PER_BLK;  // 32
  const int b = blockIdx.x / blocksPerBatch;
  const int row0 = (blockIdx.x % blocksPerBatch) * TOK_PER_BLK;
  float* hb = h + ((size_t)b * N_ + row0) * D_;
  const float* cb = ctx + b * D_;

  // ---- stage W1 panel (all threads), completed before LN1 below
  stage_w_panel(W1, Wsh, tid);

  // ---- LN1(h + ctx): each wave handles its 16 rows, 2 elems per lane
  const float c0 = cb[2 * lane], c1 = cb[2 * lane + 1];
  const float g1a = g1[2 * lane], g1b = g1[2 * lane + 1];
  const float e1a = be1[2 * lane], e1b = be1[2 * lane + 1];
#pragma unroll
  for (int t = 0; t < 16; ++t) {
    const int r = wr0 + t;
    float x0 = hb[r * D_ + 2 * lane] + c0;
    float x1 = hb[r * D_ + 2 * lane + 1] + c1;
    float s = x0 + x1, q = x0 * x0 + x1 * x1;
#pragma unroll
    for (int m = 16; m > 0; m >>= 1) {
      s += __shfl_xor(s, m, 32);
      q += __shfl_xor(q, m, 32);
    }
    float mean = s * (1.f / 64.f);
    float var = q * (1.f / 64.f) - mean * mean;
    float rs = rsqrtf(var + LNEPS);
    Aln[r * D_ + 2 * lane]     = (x0 - mean) * rs * g1a + e1a;
    Aln[r * D_ + 2 * lane + 1] = (x1 - mean) * rs * g1b + e1b;
  }
  __syncthreads();  // W1 panel staged + all Aln rows visible

  const int mrow = lane & 15;   // M row (A) / N col (B,C,D)
  const int khalf = lane >> 4;  // which K pair within a 4-step
  const int arow = wr0 + mrow;

  // ---- FFN1: [16x64] @ [64x256] per wave; B-fragments from LDS panel
  for (int nt = 0; nt < 16; ++nt) {
    v8f acc = {};
    const int n = nt * 16 + mrow;
#pragma unroll
    for (int k = 0; k < 16; ++k) {
      const int kk = k * 4 + khalf * 2;
      v2f a;  a.x = Aln[arow * D_ + kk];   a.y = Aln[arow * D_ + kk + 1];
      v2f bm; bm.x = Wsh[kk * FF_ + n];    bm.y = Wsh[(kk + 1) * FF_ + n];
      acc = __builtin_amdgcn_wmma_f32_16x16x4_f32(false, a, false, bm,
                                                  (short)0, acc, false, false);
    }
    const float bias = b1[n];
#pragma unroll
    for (int i = 0; i < 8; ++i) {
      float vv = acc[i] + bias;
      Mid[(wr0 + i + khalf * 8) * FF_ + n] = vv > 0.f ? vv : 0.f;  // relu
    }
  }
  __syncthreads();  // everyone done reading W1 panel

  // ---- restage panel with W2
  stage_w_panel(W2, Wsh, tid);
  __syncthreads();

  // ---- FFN2: [16x256] @ [256x64] per wave; residual folded into Aln
  for (int nt = 0; nt < 4; ++nt) {
    v8f acc = {};
    const int n = nt * 16 + mrow;
#pragma unroll 8
    for (int k = 0; k < 64; ++k) {
      const int kk = k * 4 + khalf * 2;
      v2f a;  a.x = Mid[arow * FF_ + kk];  a.y = Mid[arow * FF_ + kk + 1];
      v2f bm; bm.x = Wsh[kk * D_ + n];     bm.y = Wsh[(kk + 1) * D_ + n];
      acc = __builtin_amdgcn_wmma_f32_16x16x4_f32(false, a, false, bm,
                                                  (short)0, acc, false, false);
    }
    const float bias = b2[n];
#pragma unroll
    for (int i = 0; i < 8; ++i) {
      const int m = wr0 + i + khalf * 8;
      Aln[m * D_ + n] += acc[i] + bias;  // hln + ffn(hln)  (same-wave rows)
    }
  }
  // no barrier needed: LN2 below only reads this wave's own rows, and DS ops
  // from one wave execute in program order.

  // ---- LN2 and store back to h
  const float g2a = g2[2 * lane], g2b = g2[2 * lane + 1];
  const float e2a = be2[2 * lane], e2b = be2[2 * lane + 1];
#pragma unroll
  for (int t = 0; t < 16; ++t) {
    const int r = wr0 + t;
    float x0 = Aln[r * D_ + 2 * lane], x1 = Aln[r * D_ + 2 * lane + 1];
    float s = x0 + x1, q = x0 * x0 + x1 * x1;
#pragma unroll
    for (int m = 16; m > 0; m >>= 1) {
      s += __shfl_xor(s, m, 32);
      q += __shfl_xor(q, m, 32);
    }
    float mean = s * (1.f / 64.f);
    float var = q * (1.f / 64.f) - mean * mean;
    float rs = rsqrtf(var + LNEPS);
    hb[r * D_ + 2 * lane]     = (x0 - mean) * rs * g2a + e2a;
    hb[r * D_ + 2 * lane + 1] = (x1 - mean) * rs * g2b + e2b;
  }
}

// ---------------------------------------------------------------- launch
extern "C" void kernel_launch(void* const* d_in, const int* in_sizes, int n_in,
                              void* d_out, int out_size, void* d_ws,
                              size_t ws_size, hipStream_t stream) {
  (void)in_sizes; (void)n_in; (void)out_size; (void)ws_size;

  const float* x      = (const float*)d_in[0];
  const float* last   = (const float*)d_in[1];
  const float* eW     = (const float*)d_in[2];
  const float* eb     = (const float*)d_in[3];
  const float* enc_wk = (const float*)d_in[5];
  const float* W1     = (const float*)d_in[7];
  const float* b1     = (const float*)d_in[8];
  const float* W2     = (const float*)d_in[9];
  const float* b2     = (const float*)d_in[10];
  const float* g1     = (const float*)d_in[11];
  const float* be1    = (const float*)d_in[12];
  const float* g2     = (const float*)d_in[13];
  const float* be2    = (const float*)d_in[14];
  const float* dec_wk = (const float*)d_in[16];

  float* h    = (float*)d_ws;                 // [B,N,D]
  float* encM = h + (size_t)B_ * N_ * D_;     // [B,N]
  float* memM = encM + (size_t)B_ * N_;       // [B,N]
  float* ctx  = memM + (size_t)B_ * N_;       // [B,D]

  masks_kernel<<<B_, 256, 0, stream>>>(x, encM, memM);
  embed_kernel<<<(B_ * N_ * D_ + 255) / 256, 256, 0, stream>>>(x, last, eW, eb, h);

  for (int l = 0; l < L_; ++l) {
    attn_ctx_kernel<false><<<B_, 1024, 0, stream>>>(h, enc_wk + l * D_, encM,
                                                    ctx, nullptr);
    enc_ffn_kernel<<<(B_ * N_) / TOK_PER_BLK, FFN_THREADS, FFN_LDS_BYTES,
                     stream>>>(
        h, ctx, W1 + (size_t)l * D_ * FF_, b1 + l * FF_,
        W2 + (size_t)l * FF_ * D_, b2 + l * D_, g1 + l * D_, be1 + l * D_,
        g2 + l * D_, be2 + l * D_);
  }

  // Final output: last decoder layer's cross-attn weights — reduces to a
  // masked softmax of h_enc · dec.wk[5] (query token drops out of softmax).
  attn_ctx_kernel<true><<<B_, 1024, 0, stream>>>(h, dec_wk + 5 * D_, memM,
                                                 nullptr, (float*)d_out);
}